// HiddenInteractionTensor_19713899888662
// MI455X (gfx1250) — compile-verified
//
#include <hip/hip_runtime.h>
#include <hip/hip_bf16.h>

// MI455X / gfx1250 implementation.
//
// V0 = (pt ⊙ s1) @ U0^T  [P x N0],  V1 = (pt ⊙ s0) @ U1^T  [P x N1]
// Store-bound (600 MB fp32 out vs 9.6 GFLOP) but AI ~16 F/B still requires the
// matrix pipe: V_WMMA_F32_16X16X4_F32, K=32 => 8 chained WMMAs per 16x16 tile.
// Each wave computes a 16M x 64N strip: the scaled A-frags (pt rows) are
// loaded/scaled once and reused across 4 N sub-tiles, halving L1 load issue
// pressure vs a 16x16-per-wave scheme. Stores are straight-line non-temporal
// (TH_STORE_NT) so the write-once output stream does not evict pt_reps
// (6.4 MB, reused ~188x) from the 192 MB L2.

typedef __attribute__((ext_vector_type(2))) float v2f;
typedef __attribute__((ext_vector_type(8))) float v8f;

#define RANK 32

// ---------------------------------------------------------------------------
// Kernel 1: column sums.  s[0..31] = U0.sum(0), s[32..63] = U1.sum(0).
// 2 blocks x 8 waves; lane = column (128B coalesced row reads), waves stride
// rows, wave 0 sums the 8 LDS partials. Keeps this off the critical path (~250 iters/wave).
// ---------------------------------------------------------------------------
__global__ void hit_colsum_kernel(const float* __restrict__ U0, int n0,
                                  const float* __restrict__ U1, int n1,
                                  float* __restrict__ s) {
  __shared__ float partial[8][RANK];
  const float* U = blockIdx.x ? U1 : U0;
  const int n = blockIdx.x ? n1 : n0;
  float* sOut = s + (blockIdx.x ? RANK : 0);

  const int lane = threadIdx.x & 31;
  const int wave = threadIdx.x >> 5;

  float acc = 0.0f;
  for (int i = wave; i < n; i += 8) acc += U[(size_t)i * RANK + lane];
  partial[wave][lane] = acc;
  __syncthreads();

  if (wave == 0) {
    float t = 0.0f;
#pragma unroll
    for (int w = 0; w < 8; ++w) t += partial[w][lane];
    sOut[lane] = t;
  }
}

// ---------------------------------------------------------------------------
// Kernel 2: V[p][n] = sum_r (pt[p][r] * s[r]) * U[n][r]
// One wave per 16(M) x 64(N) strip = 4 sub-tiles of 16x16.
// A-frag (16x4 f32 per k-step): lane L holds row mBase+(L&15),
// cols 4k + 2*(L>>4) + {0,1} -> one global_load_b64 per step; scaled by s
// once and kept in VGPRs for all 4 sub-tiles. B-frag (4x16 f32) = U^T tile
// has the identical per-lane pattern on U's row-major [N,32] storage.
// ---------------------------------------------------------------------------
__global__ void hit_mttkrp_kernel(const float* __restrict__ pt,
                                  const float* __restrict__ U,
                                  const float* __restrict__ s,
                                  float* __restrict__ V,
                                  int P, int N, int nTiles64, int totalTiles) {
  const int lane = threadIdx.x & 31;
  const int wave = threadIdx.x >> 5;
  const int tile = blockIdx.x * (blockDim.x >> 5) + wave;
  if (tile >= totalTiles) return;            // wave-uniform: EXEC stays full

  const int mBase  = (tile / nTiles64) * 16;
  const int nBase0 = (tile % nTiles64) * 64;

  const int half16 = lane & 15;
  const int khalf  = (lane >> 4) << 1;       // 0 for lanes 0-15, 2 for 16-31

  // ---- A frags: load + scale once, reuse across 4 N sub-tiles ----
  int mRow = mBase + half16;
  if (mRow >= P) mRow = P - 1;               // only possible if P % 16 != 0
  const float* aRow = pt + (size_t)mRow * RANK + khalf;
  const float* sCol = s + khalf;
  v2f a[8];
#pragma unroll
  for (int k = 0; k < 8; ++k) {
    v2f av = *(const v2f*)(aRow + 4 * k);    // global_load_b64
    v2f sv = *(const v2f*)(sCol + 4 * k);
    a[k] = av * sv;                          // fold diag(s) into A
  }

  const int rowBase = mBase + ((lane >> 4) << 3);
  const bool mFull = (mBase + 16 <= P);      // wave-uniform

#pragma unroll
  for (int t = 0; t < 4; ++t) {
    const int nBase = nBase0 + t * 16;
    if (nBase < N) {                         // wave-uniform: skip dead tiles
      int nRow = nBase + half16;
      if (nRow >= N) nRow = N - 1;           // clamp; masked at store
      const float* bRow = U + (size_t)nRow * RANK + khalf;

      v8f c = {};
#pragma unroll
      for (int k = 0; k < 8; ++k) {
        v2f b = *(const v2f*)(bRow + 4 * k); // global_load_b64
        // 8 args: (neg_a, A, neg_b, B, c_mod, C, reuse_a, reuse_b)
        c = __builtin_amdgcn_wmma_f32_16x16x4_f32(false, a[k], false, b,
                                                  (short)0, c, false, false);
      }

      // D: lane L, vgpr v -> row rowBase+v, col nBase+(L&15).
      const int col = nBase + half16;
      float* outp = V + (size_t)rowBase * N + col;
      if (mFull && (nBase + 16 <= N)) {
        // HOT PATH (all of V0; all but 1/63 tiles of V1): straight-line
        // strided non-temporal stores, no per-element predication.
#pragma unroll
        for (int v = 0; v < 8; ++v) {
          __builtin_nontemporal_store(c[v], outp);
          outp += N;
        }
      } else {
        // Ragged edge: divergent column guard, rare.
        if (col < N) {
#pragma unroll
          for (int v = 0; v < 8; ++v) {
            if (rowBase + v < P) __builtin_nontemporal_store(c[v], outp);
            outp += N;
          }
        }
      }
    }
  }
}

// ---------------------------------------------------------------------------
// Host-side launch.  d_in = {pt_reps [P,32], U0 [N0,32], U1 [N1,32]} fp32.
// d_out = V0 flat [P*N0] then V1 flat [P*N1].  d_ws holds s0|s1 (64 floats).
// ---------------------------------------------------------------------------
extern "C" void kernel_launch(void* const* d_in, const int* in_sizes, int n_in,
                              void* d_out, int out_size, void* d_ws, size_t ws_size,
                              hipStream_t stream) {
  const float* pt = (const float*)d_in[0];
  const float* U0 = (const float*)d_in[1];
  const float* U1 = (const float*)d_in[2];
  float* out = (float*)d_out;
  float* s   = (float*)d_ws;                 // s0 at s[0..31], s1 at s[32..63]

  const int P  = in_sizes[0] / RANK;         // 50000
  const int N0 = in_sizes[1] / RANK;         // 2000
  const int N1 = in_sizes[2] / RANK;         // 1000

  hit_colsum_kernel<<<2, 256, 0, stream>>>(U0, N0, U1, N1, s);

  const int mTiles   = (P + 15) / 16;
  const int n0Tile64 = (N0 + 63) / 64;
  const int n1Tile64 = (N1 + 63) / 64;
  const int total0   = mTiles * n0Tile64;
  const int total1   = mTiles * n1Tile64;

  const int wavesPerBlock = 8;               // 256 threads = 8 wave32
  const int threads = wavesPerBlock * 32;

  // V0 uses s1 (U1 column sums); V1 uses s0.
  hit_mttkrp_kernel<<<(total0 + wavesPerBlock - 1) / wavesPerBlock, threads, 0,
                      stream>>>(pt, U0, s + RANK, out, P, N0, n0Tile64, total0);
  hit_mttkrp_kernel<<<(total1 + wavesPerBlock - 1) / wavesPerBlock, threads, 0,
                      stream>>>(pt, U1, s, out + (size_t)P * N0, P, N1,
                                n1Tile64, total1);
}